// Lorenz96Integrator_33947421507708
// MI455X (gfx1250) — compile-verified
//
#include <hip/hip_runtime.h>

#define F_FORCE 8.0f
#define DT      0.01f
#define NSTEPS  100
#define DIM     40
#define TPB     256
#define BLK_FLOATS (TPB * DIM)                  // 10240 floats = 40 KB per phase
#define VECS_PER_THREAD (BLK_FLOATS / 4 / TPB)  // 10 b128 copies per thread/phase

#define AS3 __attribute__((address_space(3)))

typedef int v4i __attribute__((vector_size(4 * sizeof(int))));
typedef float v2f __attribute__((ext_vector_type(2)));

#if defined(__AMDGCN__) &&                                                     \
    __has_builtin(__builtin_amdgcn_global_load_async_to_lds_b128) &&           \
    __has_builtin(__builtin_amdgcn_global_store_async_from_lds_b128) &&        \
    __has_builtin(__builtin_amdgcn_s_wait_asynccnt)
#define USE_ASYNC_LDS 1
#else
#define USE_ASYNC_LDS 0
#endif

// Packed fma: splat scalar * v2f + v2f  ->  v_pk_fma_f32
static __device__ __host__ __forceinline__ v2f fma2(float s, v2f b, v2f c) {
  v2f sv;
  sv.x = s;
  sv.y = s;
  return __builtin_elementwise_fma(sv, b, c);
}

// dx_i/dt = (x_{i+1} - x_{i-2}) * x_{i-1} - x_i + F   (cyclic), 2 rows packed
#define DERIV(SRC, DST)                                                        \
  _Pragma("unroll")                                                            \
  for (int i = 0; i < DIM; ++i) {                                              \
    const int ip1 = (i + 1) % DIM;                                             \
    const int im2 = (i + DIM - 2) % DIM;                                       \
    const int im1 = (i + DIM - 1) % DIM;                                       \
    DST[i] = __builtin_elementwise_fma(SRC[ip1] - SRC[im2], SRC[im1],          \
                                       F_FORCE - SRC[i]);                      \
  }

__global__ __launch_bounds__(TPB) void
lorenz96_rk38_kernel(const float* __restrict__ xin, float* __restrict__ xout) {
  __shared__ float lds[BLK_FLOATS];
  const int tid = threadIdx.x;
  // Each block handles 512 rows: phase 0 = rows [0,256), phase 1 = [256,512).
  const size_t base = (size_t)blockIdx.x * (2 * BLK_FLOATS);

  v2f x[DIM];  // .x = row (block*512 + tid), .y = row (block*512 + 256 + tid)

  // ---- Stage in: two 40KB phases through LDS (coalesced async copies) ----
#pragma unroll
  for (int ph = 0; ph < 2; ++ph) {
    const float* gsrc = xin + base + (size_t)ph * BLK_FLOATS;
#if USE_ASYNC_LDS
    {
      v4i* g4 = (v4i*)gsrc;
      AS3 v4i* l4 = (AS3 v4i*)lds;
#pragma unroll
      for (int k = 0; k < VECS_PER_THREAD; ++k) {
        const int idx = tid + k * TPB;  // lane-contiguous 16B elements
        __builtin_amdgcn_global_load_async_to_lds_b128(g4 + idx, l4 + idx, 0,
                                                       0);
      }
      __builtin_amdgcn_s_wait_asynccnt(0);
    }
#else
    {
      const float4* g4 = (const float4*)gsrc;
      float4* l4 = (float4*)lds;
#pragma unroll
      for (int k = 0; k < VECS_PER_THREAD; ++k)
        l4[tid + k * TPB] = g4[tid + k * TPB];
    }
#endif
    __syncthreads();
    {
      const float4* lrow = (const float4*)(&lds[tid * DIM]);
#pragma unroll
      for (int j = 0; j < DIM / 4; ++j) {
        float4 v = lrow[j];
        if (ph == 0) {
          x[4 * j + 0].x = v.x;
          x[4 * j + 1].x = v.y;
          x[4 * j + 2].x = v.z;
          x[4 * j + 3].x = v.w;
        } else {
          x[4 * j + 0].y = v.x;
          x[4 * j + 1].y = v.y;
          x[4 * j + 2].y = v.z;
          x[4 * j + 3].y = v.w;
        }
      }
    }
    __syncthreads();  // LDS reused by next phase / output staging
  }

  // ---- 100 RK4 (Kutta 3/8) steps, 2 rows per thread, all packed f32 ----
  v2f k1[DIM], k2[DIM], k3[DIM], a[DIM];
  for (int s = 0; s < NSTEPS; ++s) {
    DERIV(x, k1);                                    // k1 = f(x)
#pragma unroll
    for (int i = 0; i < DIM; ++i)                    // a = x + dt*k1/3
      a[i] = fma2(DT * (1.0f / 3.0f), k1[i], x[i]);

    DERIV(a, k2);                                    // k2 = f(a)
#pragma unroll
    for (int i = 0; i < DIM; ++i)                    // a = x + dt*(k2 - k1/3)
      a[i] = fma2(DT, k2[i], fma2(-DT * (1.0f / 3.0f), k1[i], x[i]));

    DERIV(a, k3);                                    // k3 = f(a)
#pragma unroll
    for (int i = 0; i < DIM; ++i)                    // a = x + dt*(k1-k2+k3)
      a[i] = fma2(DT, (k1[i] - k2[i]) + k3[i], x[i]);
#pragma unroll
    for (int i = 0; i < DIM; ++i)                    // k1 <- k1 + 3*(k2+k3)
      k1[i] = fma2(3.0f, k2[i] + k3[i], k1[i]);

    DERIV(a, k2);                                    // k4 = f(a), reuse k2 regs
#pragma unroll
    for (int i = 0; i < DIM; ++i)                    // x += dt*(sum)/8
      x[i] = fma2(DT * (1.0f / 8.0f), k1[i] + k2[i], x[i]);
  }

  // ---- Stage out: two 40KB phases through LDS (coalesced async stores) ----
#pragma unroll
  for (int ph = 0; ph < 2; ++ph) {
    {
      float4* lrow = (float4*)(&lds[tid * DIM]);
#pragma unroll
      for (int j = 0; j < DIM / 4; ++j) {
        float4 v;
        if (ph == 0) {
          v.x = x[4 * j + 0].x;
          v.y = x[4 * j + 1].x;
          v.z = x[4 * j + 2].x;
          v.w = x[4 * j + 3].x;
        } else {
          v.x = x[4 * j + 0].y;
          v.y = x[4 * j + 1].y;
          v.z = x[4 * j + 2].y;
          v.w = x[4 * j + 3].y;
        }
        lrow[j] = v;
      }
    }
    __syncthreads();
    float* gdst = xout + base + (size_t)ph * BLK_FLOATS;
#if USE_ASYNC_LDS
    {
      v4i* g4 = (v4i*)gdst;
      AS3 v4i* l4 = (AS3 v4i*)lds;
#pragma unroll
      for (int k = 0; k < VECS_PER_THREAD; ++k) {
        const int idx = tid + k * TPB;
        __builtin_amdgcn_global_store_async_from_lds_b128(g4 + idx, l4 + idx, 0,
                                                          0);
      }
      __builtin_amdgcn_s_wait_asynccnt(0);  // stores must finish before LDS reuse
    }
#else
    {
      float4* g4 = (float4*)gdst;
      const float4* l4 = (const float4*)lds;
#pragma unroll
      for (int k = 0; k < VECS_PER_THREAD; ++k)
        g4[tid + k * TPB] = l4[tid + k * TPB];
    }
#endif
    __syncthreads();
  }
}

extern "C" void kernel_launch(void* const* d_in, const int* in_sizes, int n_in,
                              void* d_out, int out_size, void* d_ws,
                              size_t ws_size, hipStream_t stream) {
  const float* x = (const float*)d_in[0];
  float* out = (float*)d_out;
  const int batch = in_sizes[0] / DIM;                     // 262144
  const int rows_per_block = 2 * TPB;                      // 512
  const int blocks = (batch + rows_per_block - 1) / rows_per_block;  // 512
  lorenz96_rk38_kernel<<<blocks, TPB, 0, stream>>>(x, out);
}